// LocalOptLoss_4466765988626
// MI455X (gfx1250) — compile-verified
//
#include <hip/hip_runtime.h>

typedef __attribute__((ext_vector_type(2))) float v2f;
typedef __attribute__((ext_vector_type(8))) float v8f;

#define TILE_LD 17

// ---------------- LDS layout (float offsets) ----------------
// Weights (row-major, row stride = cols+1 to dodge bank conflicts)
constexpr int oWT1   = 0;      // 32x16 ld17 (544)
constexpr int oWT2   = 544;    // 16x32 ld33 (528)
constexpr int oWtau1 = 1072;   // 32x16 ld17
constexpr int oWtau2 = 1616;   // 16x32 ld33
constexpr int oWh1   = 2144;   // 32x16 ld17
constexpr int oWh2   = 2688;   // 8x32 used, 16x33 alloc (rows 8..15 zeroed)
constexpr int oWpsi1 = 3216;   // 32x24 ld25 (800)
constexpr int oWpsi2 = 4016;   // 16x32 ld33
constexpr int oWf1   = 4544;   // 32x16 ld17
constexpr int oWf2   = 5088;   // 16x32 ld33
constexpr int oWP    = 5616;   // 16x16 ld17 (272)
constexpr int obT1   = 5888;
constexpr int obtau1 = 5920;
constexpr int obh1   = 5952;
constexpr int obpsi1 = 5984;
constexpr int obf1   = 6016;
// Per-wave tiles (ld = TILE_LD = 17)
constexpr int oX     = 6048;   // X -> diff          (16 rows)
constexpr int oE     = 6320;   // E                  (16)
constexpr int oT16a  = 6592;   // Z / q1 / je        (16)
constexpr int oT16b  = 6864;   // Xh / q2,q3,q4      (16)
constexpr int oT16c  = 7136;   // PHI / s            (16)
constexpr int oCC    = 7408;   // 32 rows alloc: [0:16)=TX -> v/term1/RES ;
                               // [16:24)=Yh -> w ; [24:32) junk rows from M=8 stores
constexpr int oS0    = 7952;   // d4 -> df           (32)
constexpr int oS1    = 8496;   // d5                 (32)
constexpr int oS2    = 9040;   // d6                 (32)
constexpr int oS3    = 9584;   // c6                 (32)
constexpr int oS4    = 10128;  // U scratch          (32)
constexpr int oS5    = 10672;  // V/m/tmp scratch    (32)
constexpr int oNRM   = 11216;  // 16 column scalars
constexpr int LDS_TOTAL = 11232;  // ~44.9 KB

// D(Mx16) = W(MxK) @ S(Kx16) [+ bias], via V_WMMA_F32_16X16X4_F32.
// A frag: lane L, reg r -> A[M = rb + L%16, K = kc + (L/16)*2 + r]
// B frag: lane L, reg r -> B[K = kc + (L/16)*2 + r, N = L%16]
// C/D   : lane L, reg r -> D[M = rb + r + (L/16)*8, N = L%16]
// NOTE: stores a full ceil(M/16)*16-row block unconditionally (no divergent
// guards); destination regions are sized accordingly. All params compile-time
// so no runtime branches survive around the WMMA cores.
template <int M, int K, bool TRANS, bool BIAS>
__device__ __forceinline__ void wave_matmul(
    float* __restrict__ D,
    const float* __restrict__ W, int ldW,
    const float* __restrict__ S,
    const float* __restrict__ bias,
    int lane)
{
  const int mloc = lane & 15;
  const int koff = (lane >> 4) << 1;  // 0 or 2
  const int nB   = lane & 15;
#pragma unroll
  for (int rb = 0; rb < M; rb += 16) {
    v8f acc = {0.f, 0.f, 0.f, 0.f, 0.f, 0.f, 0.f, 0.f};
#pragma unroll
    for (int kc = 0; kc < K; kc += 4) {
      const int k0 = kc + koff;
      v2f a, b;
      if (!TRANS) {
        a.x = W[(rb + mloc) * ldW + k0];
        a.y = W[(rb + mloc) * ldW + k0 + 1];
      } else {
        a.x = W[(k0    ) * ldW + rb + mloc];
        a.y = W[(k0 + 1) * ldW + rb + mloc];
      }
      b.x = S[(k0    ) * TILE_LD + nB];
      b.y = S[(k0 + 1) * TILE_LD + nB];
      acc = __builtin_amdgcn_wmma_f32_16x16x4_f32(false, a, false, b,
                                                  (short)0, acc, false, false);
    }
    const int mBase = rb + ((lane >> 4) << 3);
#pragma unroll
    for (int r = 0; r < 8; ++r) {
      float v = acc[r];
      if (BIAS) v += bias[mBase + r];
      D[(mBase + r) * TILE_LD + nB] = v;
    }
  }
}

__device__ __forceinline__ void load_w(float* dst, int ld,
                                       const float* __restrict__ src,
                                       int rows, int cols, int lane) {
  for (int idx = lane; idx < rows * cols; idx += 32) {
    int r = idx / cols, c = idx - r * cols;
    dst[r * ld + c] = src[idx];
  }
}

// Branch-free tanh: copysign((1 - e^{-2|x|}) / (1 + e^{-2|x|}), x).
// e^{-2|x|} in (0,1] -> no overflow/NaN path, lowers to v_exp_f32 + few VALU.
__device__ __forceinline__ float fast_tanh(float x) {
  float t = __expf(-2.f * fabsf(x));
  float r = __fdividef(1.f - t, 1.f + t);
  return copysignf(r, x);
}

__device__ __forceinline__ void ew_tanh(float* T, int M, int lane) {
  for (int idx = lane; idx < M * 16; idx += 32) {
    int r = idx >> 4, c = idx & 15;
    T[r * TILE_LD + c] = fast_tanh(T[r * TILE_LD + c]);
  }
}

#define EW_LOOP(M) for (int idx = lane; idx < (M) * 16; idx += 32)
#define EW_AT(P) (P)[(idx >> 4) * TILE_LD + (idx & 15)]

__global__ __launch_bounds__(32)
void localopt_kernel(
    const float* __restrict__ x_batch, const float* __restrict__ e_batch,
    const float* __restrict__ Wf1,  const float* __restrict__ bf1,  const float* __restrict__ Wf2,
    const float* __restrict__ Wh1,  const float* __restrict__ bh1,  const float* __restrict__ Wh2,
    const float* __restrict__ WT1,  const float* __restrict__ bT1,  const float* __restrict__ WT2,
    const float* __restrict__ Wtau1,const float* __restrict__ btau1,const float* __restrict__ Wtau2,
    const float* __restrict__ Wpsi1,const float* __restrict__ bpsi1,const float* __restrict__ Wpsi2,
    const float* __restrict__ WP,
    float* __restrict__ partial)
{
  __shared__ float lds[LDS_TOTAL];
  const int lane = threadIdx.x;

  // ---- stage weights into padded LDS ----
  load_w(lds + oWT1,   17, WT1,   32, 16, lane);
  load_w(lds + oWT2,   33, WT2,   16, 32, lane);
  load_w(lds + oWtau1, 17, Wtau1, 32, 16, lane);
  load_w(lds + oWtau2, 33, Wtau2, 16, 32, lane);
  load_w(lds + oWh1,   17, Wh1,   32, 16, lane);
  load_w(lds + oWh2,   33, Wh2,    8, 32, lane);
  for (int idx = lane; idx < 8 * 33; idx += 32) lds[oWh2 + 8 * 33 + idx] = 0.f;
  load_w(lds + oWpsi1, 25, Wpsi1, 32, 24, lane);
  load_w(lds + oWpsi2, 33, Wpsi2, 16, 32, lane);
  load_w(lds + oWf1,   17, Wf1,   32, 16, lane);
  load_w(lds + oWf2,   33, Wf2,   16, 32, lane);
  load_w(lds + oWP,    17, WP,    16, 16, lane);
  lds[obT1 + lane] = bT1[lane];     lds[obtau1 + lane] = btau1[lane];
  lds[obh1 + lane] = bh1[lane];     lds[obpsi1 + lane] = bpsi1[lane];
  lds[obf1 + lane] = bf1[lane];

  // ---- load 16-element batch tile, transposed to (feature x batch) ----
  const int b0 = blockIdx.x * 16;
  for (int idx = lane; idx < 256; idx += 32) {
    int n = idx >> 4, j = idx & 15;
    lds[oX + n * TILE_LD + j] = x_batch[(size_t)(b0 + j) * 16 + n];
    lds[oE + n * TILE_LD + j] = e_batch[(size_t)(b0 + j) * 16 + n];
  }
  __syncthreads();

  float* X   = lds + oX;   float* E  = lds + oE;
  float* Z   = lds + oT16a;              // Z / q1 / je
  float* Xh  = lds + oT16b;              // x_hat, later q2/q3/q4 staging
  float* PHI = lds + oT16c;              // phi, later s
  float* TX  = lds + oCC;                // Tx, later v/term1/RES
  float* YH  = lds + oCC + 16 * TILE_LD; // y_hat (8 rows + junk), later w
  float* D4  = lds + oS0;  float* D5 = lds + oS1; float* D6 = lds + oS2;
  float* C6  = lds + oS3;  float* U  = lds + oS4; float* V  = lds + oS5;
  float* NRM = lds + oNRM;

  // 1) z = T(x) + e
  wave_matmul<32, 16, false, true >(U, lds + oWT1, 17, X, lds + obT1, lane); __syncthreads();
  ew_tanh(U, 32, lane); __syncthreads();
  wave_matmul<16, 32, false, false>(Z, lds + oWT2, 33, U, nullptr, lane); __syncthreads();
  EW_LOOP(16) EW_AT(Z) += EW_AT(E);
  __syncthreads();

  // 2) x_hat = tau(z)
  wave_matmul<32, 16, false, true >(U, lds + oWtau1, 17, Z, lds + obtau1, lane); __syncthreads();
  ew_tanh(U, 32, lane); __syncthreads();
  wave_matmul<16, 32, false, false>(Xh, lds + oWtau2, 33, U, nullptr, lane); __syncthreads();

  // 3) Tx = T(x_hat)
  wave_matmul<32, 16, false, true >(U, lds + oWT1, 17, Xh, lds + obT1, lane); __syncthreads();
  ew_tanh(U, 32, lane); __syncthreads();
  wave_matmul<16, 32, false, false>(TX, lds + oWT2, 33, U, nullptr, lane); __syncthreads();

  // 4) y_hat = h(x_hat); d4 = h' inner diag
  wave_matmul<32, 16, false, true >(U, lds + oWh1, 17, Xh, lds + obh1, lane); __syncthreads();
  ew_tanh(U, 32, lane); __syncthreads();
  EW_LOOP(32) { float t = EW_AT(U); EW_AT(D4) = 1.f - t * t; }
  __syncthreads();
  wave_matmul< 8, 32, false, false>(YH, lds + oWh2, 33, U, nullptr, lane); __syncthreads();

  // 5) phi = psi([Tx; y_hat]); d5 = psi' inner diag
  wave_matmul<32, 24, false, true >(U, lds + oWpsi1, 25, TX, lds + obpsi1, lane); __syncthreads();
  ew_tanh(U, 32, lane); __syncthreads();
  EW_LOOP(32) { float t = EW_AT(U); EW_AT(D5) = 1.f - t * t; }
  __syncthreads();
  wave_matmul<16, 32, false, false>(PHI, lds + oWpsi2, 33, U, nullptr, lane); __syncthreads();

  // 6) tau internals at Tx: t6, d6, c6 = -2 t6 d6 * (Wtau1 @ phi)
  wave_matmul<32, 16, false, true >(U, lds + oWtau1, 17, TX, lds + obtau1, lane); __syncthreads();
  ew_tanh(U, 32, lane); __syncthreads();
  EW_LOOP(32) { float t = EW_AT(U); EW_AT(D6) = 1.f - t * t; }
  __syncthreads();
  wave_matmul<32, 16, false, false>(V, lds + oWtau1, 17, PHI, nullptr, lane); __syncthreads();
  EW_LOOP(32) EW_AT(C6) = -2.f * EW_AT(U) * EW_AT(D6) * EW_AT(V);
  __syncthreads();

  // 7) term1 = P_inv @ (Jh^T @ ((Jh @ diff)/R))
  EW_LOOP(16) EW_AT(X) -= EW_AT(Xh);                          // diff -> X
  __syncthreads();
  wave_matmul<32, 16, false, false>(U, lds + oWh1, 17, X, nullptr, lane); __syncthreads();
  EW_LOOP(32) EW_AT(U) *= EW_AT(D4);
  __syncthreads();
  wave_matmul< 8, 32, false, false>(YH, lds + oWh2, 33, U, nullptr, lane); __syncthreads(); // w = Jh@diff
  EW_LOOP(8) EW_AT(YH) *= 10.f;                               // /R, R=0.1
  __syncthreads();
  wave_matmul<32,  8, true,  false>(V, lds + oWh2, 33, YH, nullptr, lane); __syncthreads(); // Wh2^T @ w
  EW_LOOP(32) EW_AT(V) *= EW_AT(D4);
  __syncthreads();
  wave_matmul<16, 32, true,  false>(TX, lds + oWh1, 17, V, nullptr, lane); __syncthreads(); // v -> TX slot
  wave_matmul<16, 16, false, false>(PHI, lds + oWP, 17, Xh, nullptr, lane); __syncthreads();
  ew_tanh(PHI, 16, lane); __syncthreads();                    // s
  if (lane < 16) {                                            // per-column s.v
    float d = 0.f;
    for (int i = 0; i < 16; ++i) d += PHI[i * TILE_LD + lane] * TX[i * TILE_LD + lane];
    NRM[lane] = d;
  }
  __syncthreads();
  EW_LOOP(16) EW_AT(TX) += EW_AT(PHI) * NRM[idx & 15];        // RES = term1
  __syncthreads();

  // 8a) df = f' inner diag at x_hat (reuse D4 slot)
  wave_matmul<32, 16, false, true >(U, lds + oWf1, 17, Xh, lds + obf1, lane); __syncthreads();
  ew_tanh(U, 32, lane); __syncthreads();
  EW_LOOP(32) { float t = EW_AT(U); EW_AT(D4) = 1.f - t * t; }
  __syncthreads();
  // 8b) m = Wtau1@e ; q1 = Jtau@e
  wave_matmul<32, 16, false, false>(U, lds + oWtau1, 17, E, nullptr, lane); __syncthreads(); // m
  EW_LOOP(32) EW_AT(V) = EW_AT(D6) * EW_AT(U);
  __syncthreads();
  wave_matmul<16, 32, false, false>(Z, lds + oWtau2, 33, V, nullptr, lane); __syncthreads(); // q1
  // q2 = Jf @ q1 ; RES += q2
  wave_matmul<32, 16, false, false>(V, lds + oWf1, 17, Z, nullptr, lane); __syncthreads();
  EW_LOOP(32) EW_AT(V) *= EW_AT(D4);
  __syncthreads();
  wave_matmul<16, 32, false, false>(Xh, lds + oWf2, 33, V, nullptr, lane); __syncthreads();
  EW_LOOP(16) EW_AT(TX) += EW_AT(Xh);
  __syncthreads();
  // q4 = Hessian-term @ e ; RES -= q4   (m still in U)
  EW_LOOP(32) EW_AT(V) = EW_AT(C6) * EW_AT(U);
  __syncthreads();
  wave_matmul<16, 32, false, false>(Xh, lds + oWtau2, 33, V, nullptr, lane); __syncthreads();
  EW_LOOP(16) EW_AT(TX) -= EW_AT(Xh);
  __syncthreads();
  // q3 = Jtau @ (Jphi @ e) ; RES -= q3
  wave_matmul<32, 16, false, false>(U, lds + oWpsi1, 25, E, nullptr, lane); __syncthreads(); // Wpsi1[:,:NZ]@e
  EW_LOOP(32) EW_AT(U) *= EW_AT(D5);
  __syncthreads();
  wave_matmul<16, 32, false, false>(Z, lds + oWpsi2, 33, U, nullptr, lane); __syncthreads(); // Jphi@e
  wave_matmul<32, 16, false, false>(U, lds + oWtau1, 17, Z, nullptr, lane); __syncthreads();
  EW_LOOP(32) EW_AT(U) *= EW_AT(D6);
  __syncthreads();
  wave_matmul<16, 32, false, false>(Xh, lds + oWtau2, 33, U, nullptr, lane); __syncthreads();
  EW_LOOP(16) EW_AT(TX) -= EW_AT(Xh);
  __syncthreads();

  // per-column 2-norms, then per-tile sum
  if (lane < 16) {
    float d = 0.f;
    for (int i = 0; i < 16; ++i) { float t = TX[i * TILE_LD + lane]; d += t * t; }
    NRM[lane] = sqrtf(d);
  }
  __syncthreads();
  if (lane == 0) {
    float t = 0.f;
    for (int j = 0; j < 16; ++j) t += NRM[j];
    partial[blockIdx.x] = t;
  }
}

__global__ __launch_bounds__(256)
void reduce_kernel(const float* __restrict__ p, float* __restrict__ out,
                   int n, float invB)
{
  __shared__ float buf[256];
  float s = 0.f;
  for (int i = threadIdx.x; i < n; i += 256) s += p[i];
  buf[threadIdx.x] = s;
  __syncthreads();
  for (int off = 128; off > 0; off >>= 1) {
    if (threadIdx.x < (unsigned)off) buf[threadIdx.x] += buf[threadIdx.x + off];
    __syncthreads();
  }
  if (threadIdx.x == 0) out[0] = buf[0] * invB;
}

extern "C" void kernel_launch(void* const* d_in, const int* in_sizes, int n_in,
                              void* d_out, int out_size, void* d_ws, size_t ws_size,
                              hipStream_t stream) {
  const float* x     = (const float*)d_in[0];
  const float* e     = (const float*)d_in[1];
  const float* Wf1   = (const float*)d_in[2];
  const float* bf1   = (const float*)d_in[3];
  const float* Wf2   = (const float*)d_in[4];
  const float* Wh1   = (const float*)d_in[5];
  const float* bh1   = (const float*)d_in[6];
  const float* Wh2   = (const float*)d_in[7];
  const float* WT1   = (const float*)d_in[8];
  const float* bT1   = (const float*)d_in[9];
  const float* WT2   = (const float*)d_in[10];
  const float* Wtau1 = (const float*)d_in[11];
  const float* btau1 = (const float*)d_in[12];
  const float* Wtau2 = (const float*)d_in[13];
  const float* Wpsi1 = (const float*)d_in[14];
  const float* bpsi1 = (const float*)d_in[15];
  const float* Wpsi2 = (const float*)d_in[16];
  const float* WP    = (const float*)d_in[17];

  const int B     = in_sizes[0] / 16;   // N = 16
  const int tiles = B / 16;             // 16 batch elements per wave
  float* partial  = (float*)d_ws;

  localopt_kernel<<<tiles, 32, 0, stream>>>(
      x, e, Wf1, bf1, Wf2, Wh1, bh1, Wh2, WT1, bT1, WT2,
      Wtau1, btau1, Wtau2, Wpsi1, bpsi1, Wpsi2, WP, partial);

  reduce_kernel<<<1, 256, 0, stream>>>(partial, (float*)d_out, tiles,
                                       1.0f / (float)B);
}